// PointPillarsVoxelization_50148038148207
// MI455X (gfx1250) — compile-verified
//
#include <hip/hip_runtime.h>
#include <hip/hip_bf16.h>

// PointPillars voxelization for MI455X (gfx1250, wave32).
// Counting-sort over a dense cell grid (432*496*1 = 214,272 cells for the
// reference inputs; CAP below bounds the supported grid).
//
// Workspace layout (needs ~12.1 MB):
//   [0,            CAP*4)         counts   (u32 per cell)
//   [CAP*4,        2*CAP*4)       fill     (u32 per cell, scatter cursors)
//   [2*CAP*4,      4*CAP*4)       scan     (u64 per cell: lo=cellStart, hi=voxelId)
//   [4*CAP*4,      4*CAP*4+NB0*8) blockSums (u64)
//   [.. +256 pad,  +N*4)          sorted_idx (point ids, grouped by cell)

#define CAP        (1 << 18)               // max cells supported (>= 214,272)
#define SCAN_BLK   256
#define EPT        4                       // elements per thread in level-0 scan
#define SCAN_TILE  (SCAN_BLK * EPT)        // 1024 cells per block
#define NB0        (CAP / SCAN_TILE)       // 256 level-0 blocks
#define DPOINT     8                       // xyz + 5 feats

__device__ __forceinline__ void compute_grid(const float* vs, const float* rmin,
                                             const float* rmax,
                                             int& gx, int& gy, int& gz) {
  // jnp.round == round-to-nearest-even == rintf in default mode
  gx = (int)rintf((rmax[0] - rmin[0]) / vs[0]);
  gy = (int)rintf((rmax[1] - rmin[1]) / vs[1]);
  gz = (int)rintf((rmax[2] - rmin[2]) / vs[2]);
}

__device__ __forceinline__ long long point_cell(const float* pf, int i,
                                                const float* vs, const float* rmin,
                                                const float* rmax,
                                                int gx, int gy, int gz) {
  // one b128 load grabs xyz (+1st feat)
  float4 a = *(const float4*)(pf + (size_t)i * DPOINT);
  int vx = (int)floorf((a.x - rmin[0]) / vs[0]);
  int vy = (int)floorf((a.y - rmin[1]) / vs[1]);
  int vz = (int)floorf((a.z - rmin[2]) / vs[2]);
  bool valid = (vx >= 0) & (vx < gx) & (vy >= 0) & (vy < gy) & (vz >= 0) & (vz < gz);
  if (!valid) return -1;
  long long lin = ((long long)vz * gy + vy) * gx + vx;
  if (lin >= CAP) return -1;                // grid larger than supported cap
  return lin;
}

// ---------------------------------------------------------------- zero fill
__global__ void k_zero(float* __restrict__ out, long long n) {
  long long i = (long long)blockIdx.x * blockDim.x + threadIdx.x;
  long long stride = (long long)gridDim.x * blockDim.x;
  long long n4 = n >> 2;
  float4 z = make_float4(0.f, 0.f, 0.f, 0.f);
  for (long long j = i; j < n4; j += stride) ((float4*)out)[j] = z;   // b128 stores
  for (long long j = (n4 << 2) + i; j < n; j += stride) out[j] = 0.f; // tail
}

// ------------------------------------------------------------- cell histogram
__global__ void k_count(const float* __restrict__ pf, const float* __restrict__ vs,
                        const float* __restrict__ rmin, const float* __restrict__ rmax,
                        int N, unsigned int* __restrict__ counts) {
  int i = blockIdx.x * blockDim.x + threadIdx.x;
  if (i >= N) return;
  int gx, gy, gz;
  compute_grid(vs, rmin, rmax, gx, gy, gz);
  long long lin = point_cell(pf, i, vs, rmin, rmax, gx, gy, gz);
  if (lin >= 0) atomicAdd(&counts[lin], 1u);
}

// --------------------------- level-0 scan: packed (occupied<<32)|count, per 1024
__global__ void __launch_bounds__(SCAN_BLK)
k_scan1(const unsigned int* __restrict__ counts,
        unsigned long long* __restrict__ scan,
        unsigned long long* __restrict__ blockSums) {
  __shared__ unsigned long long s[SCAN_BLK];
  int b = blockIdx.x, t = threadIdx.x;
  int base = b * SCAN_TILE + t * EPT;
  uint4 c4 = ((const uint4*)counts)[base >> 2];   // b128 load of 4 cell counts
  unsigned int cc[EPT] = {c4.x, c4.y, c4.z, c4.w};
  unsigned long long pre[EPT];
  unsigned long long tsum = 0;
  #pragma unroll
  for (int j = 0; j < EPT; ++j) {
    unsigned long long pv =
        (unsigned long long)cc[j] | ((unsigned long long)(cc[j] != 0u) << 32);
    pre[j] = tsum;
    tsum += pv;
  }
  s[t] = tsum;
  __syncthreads();
  #pragma unroll
  for (int off = 1; off < SCAN_BLK; off <<= 1) {   // Hillis-Steele in LDS
    unsigned long long x = (t >= off) ? s[t - off] : 0ull;
    __syncthreads();
    s[t] += x;
    __syncthreads();
  }
  unsigned long long texcl = s[t] - tsum;
  #pragma unroll
  for (int j = 0; j < EPT; ++j) scan[base + j] = texcl + pre[j];
  if (t == SCAN_BLK - 1) blockSums[b] = s[t];
}

// ----------------------------------------- level-1 scan of NB0 block sums
__global__ void __launch_bounds__(NB0)
k_scan2(unsigned long long* __restrict__ blockSums) {
  __shared__ unsigned long long s[NB0];
  int t = threadIdx.x;
  unsigned long long v = blockSums[t];
  s[t] = v;
  __syncthreads();
  #pragma unroll
  for (int off = 1; off < NB0; off <<= 1) {
    unsigned long long x = (t >= off) ? s[t - off] : 0ull;
    __syncthreads();
    s[t] += x;
    __syncthreads();
  }
  blockSums[t] = s[t] - v;   // exclusive
}

// ----------------------------------------- add block offsets -> global scan
__global__ void k_scan3(unsigned long long* __restrict__ scan,
                        const unsigned long long* __restrict__ blockSums) {
  int i = blockIdx.x * blockDim.x + threadIdx.x;
  if (i < CAP) scan[i] += blockSums[i / SCAN_TILE];
}

// ------------------------------------- counting-sort scatter of point ids
__global__ void k_scatter(const float* __restrict__ pf, const float* __restrict__ vs,
                          const float* __restrict__ rmin, const float* __restrict__ rmax,
                          int N, const unsigned long long* __restrict__ scan,
                          unsigned int* __restrict__ fill, int* __restrict__ sorted_idx) {
  int i = blockIdx.x * blockDim.x + threadIdx.x;
  if (i >= N) return;
  int gx, gy, gz;
  compute_grid(vs, rmin, rmax, gx, gy, gz);
  long long lin = point_cell(pf, i, vs, rmin, rmax, gx, gy, gz);
  if (lin < 0) return;
  unsigned int start = (unsigned int)(scan[lin] & 0xffffffffull);
  unsigned int r = atomicAdd(&fill[lin], 1u);
  sorted_idx[start + r] = i;
}

// --------------- per-cell: async-stage ids to LDS, sort, gather rows b128
#define MAX_SEG 1024
__global__ void __launch_bounds__(64)
k_emit(const float* __restrict__ pf, const float* __restrict__ vs,
       const float* __restrict__ rmin, const float* __restrict__ rmax,
       const unsigned int* __restrict__ counts,
       const unsigned long long* __restrict__ scan,
       const int* __restrict__ sorted_idx,
       const int* __restrict__ p_maxp, const int* __restrict__ p_maxv,
       float* __restrict__ out) {
  __shared__ int lds[MAX_SEG];   // single static LDS alloc -> offset 0
  int cell = blockIdx.x;
  int gx, gy, gz;
  compute_grid(vs, rmin, rmax, gx, gy, gz);
  long long n_cells = (long long)gx * gy * gz;
  if (n_cells > CAP) n_cells = CAP;
  if (cell >= n_cells) return;
  unsigned int cnt = counts[cell];
  if (cnt == 0u) return;
  unsigned long long sv = scan[cell];
  int vid = (int)(sv >> 32);
  int max_voxels = *p_maxv;
  int maxp = *p_maxp;
  if (vid >= max_voxels) return;
  unsigned int start = (unsigned int)(sv & 0xffffffffull);
  int m = (cnt > MAX_SEG) ? MAX_SEG : (int)cnt;

  // CDNA5 async global->LDS staging of this cell's point-id segment.
  // LDS dest address per lane is the byte offset of lds[j] (base 0).
  for (int j = threadIdx.x; j < m; j += blockDim.x) {
    unsigned int ldsOff = (unsigned int)(j * 4);
    const int* gaddr = sorted_idx + start + j;
    asm volatile("global_load_async_to_lds_b32 %0, %1, off"
                 :: "v"(ldsOff), "v"(gaddr) : "memory");
  }
  asm volatile("s_wait_asynccnt 0" ::: "memory");
  __syncthreads();

  // Stable order = ascending original point index (reference's stable sort).
  if (threadIdx.x == 0) {
    for (int i = 1; i < m; ++i) {
      int key = lds[i];
      int j = i - 1;
      while (j >= 0 && lds[j] > key) { lds[j + 1] = lds[j]; --j; }
      lds[j + 1] = key;
    }
  }
  __syncthreads();

  int npts = ((int)cnt < maxp) ? (int)cnt : maxp;
  if (npts > m) npts = m;

  float* outVox    = out;
  float* outCoords = out + (size_t)max_voxels * (size_t)maxp * DPOINT;
  float* outNum    = outCoords + (size_t)max_voxels * 3;

  for (int r = threadIdx.x; r < npts; r += blockDim.x) {
    int p = lds[r];
    const float4* src = (const float4*)(pf + (size_t)p * DPOINT);
    float4 a = src[0];                       // global_load_b128 x2
    float4 b = src[1];
    float4* dst = (float4*)(outVox + ((size_t)vid * maxp + r) * DPOINT);
    dst[0] = a;                              // global_store_b128 x2
    dst[1] = b;
  }
  if (threadIdx.x == 0) {
    int vx = cell % gx;
    int vy = (cell / gx) % gy;
    int vz = cell / (gx * gy);
    outCoords[(size_t)vid * 3 + 0] = (float)vz;   // reference order [z, y, x]
    outCoords[(size_t)vid * 3 + 1] = (float)vy;
    outCoords[(size_t)vid * 3 + 2] = (float)vx;
    outNum[vid] = (float)npts;
  }
}

extern "C" void kernel_launch(void* const* d_in, const int* in_sizes, int n_in,
                              void* d_out, int out_size, void* d_ws, size_t ws_size,
                              hipStream_t stream) {
  const float* pf   = (const float*)d_in[0];   // [N, 8]
  const float* vs   = (const float*)d_in[1];   // [3]
  const float* rmin = (const float*)d_in[2];   // [3]
  const float* rmax = (const float*)d_in[3];   // [3]
  const int*  maxp  = (const int*)d_in[4];     // scalar 64
  const int*  maxv  = (const int*)d_in[5];     // scalar 60000
  int N = in_sizes[0] / DPOINT;
  float* out = (float*)d_out;

  char* ws = (char*)d_ws;
  unsigned int*       counts = (unsigned int*)(ws);
  unsigned int*       fill   = (unsigned int*)(ws + (size_t)CAP * 4);
  unsigned long long* scan   = (unsigned long long*)(ws + (size_t)2 * CAP * 4);
  unsigned long long* bsums  = (unsigned long long*)(ws + (size_t)4 * CAP * 4);
  int*                sorted = (int*)(ws + (size_t)4 * CAP * 4 + (size_t)NB0 * 8 + 256);

  // 1) zero per-cell counters (counts+fill are contiguous) and the full output
  k_zero<<<512, 256, 0, stream>>>((float*)counts, (long long)2 * CAP);
  k_zero<<<2048, 256, 0, stream>>>(out, (long long)out_size);
  // 2) histogram
  k_count<<<(N + 255) / 256, 256, 0, stream>>>(pf, vs, rmin, rmax, N, counts);
  // 3) packed exclusive scan: lo = cell segment start, hi = voxel id
  k_scan1<<<NB0, SCAN_BLK, 0, stream>>>(counts, scan, bsums);
  k_scan2<<<1, NB0, 0, stream>>>(bsums);
  k_scan3<<<CAP / 256, 256, 0, stream>>>(scan, bsums);
  // 4) counting-sort scatter of point ids
  k_scatter<<<(N + 255) / 256, 256, 0, stream>>>(pf, vs, rmin, rmax, N, scan, fill, sorted);
  // 5) per-cell emit (async-LDS stage + stable sort + b128 gather)
  k_emit<<<CAP, 64, 0, stream>>>(pf, vs, rmin, rmax, counts, scan, sorted, maxp, maxv, out);
}